// SoGA_67276367724779
// MI455X (gfx1250) — compile-verified
//
#include <hip/hip_runtime.h>

// SoGA forward for MI455X (gfx1250, wave32).
// B=32, C=1024, N=512, OUT=256. All fp32 (the sq[n]+sq[m]-2G cancellation
// forbids low-precision matmul inputs; diagonal must come out ~0).
// Gram (17.2 GFLOP) and S*S (8.6 GFLOP) GEMMs: V_WMMA_F32_16X16X4_F32 with
// 2x4 register blocking per wave (8 accumulators, 12 loads / 8 WMMAs per
// K-step -> ~10.7 flops per L2 byte instead of 4).
// Final [B x 262144] @ [262144 x 256] GEMM is HBM-bound on W (256 MB,
// ~11.5 us at 23.3 TB/s) -> split-K VALU kernel reading W exactly once,
// with global_prefetch on the W stream.

typedef __attribute__((ext_vector_type(2))) float v2f;
typedef __attribute__((ext_vector_type(8))) float v8f;

#define NB   32
#define NC   1024
#define NN   512
#define NOUT 256
#define NSPLIT 256
#define EPS_SQRT 1e-12f
#define EPS_NORM 1e-12f

// ---------------------------------------------------------------------------
// Kernel 1: S[b] = xo[b]^T xo[b]   (S[n,m] = sum_c xo[c,n]*xo[c,m])
// One wave per 32x64 output block (2x4 tiles of 16x16).
// A[M,k] = xo[k, tm+M], B[k,N] = xo[k, tn+N]: both are row-reads of xo,
// lanes 0..15 hit consecutive addresses (coalesced), all 12 fragment loads
// per K-step come from two row pointers with constant byte offsets.
// FP32 WMMA layout (ISA 7.12.2):
//   A 16x4: v0 = K=0 (lanes 0-15) / K=2 (lanes 16-31); v1 = K=1 / K=3
//   B 4x16: mirrored; C/D: vgpr r -> row r (lanes 0-15) / r+8 (lanes 16-31)
// ---------------------------------------------------------------------------
__global__ __launch_bounds__(256) void soga_gram(const float* __restrict__ xo,
                                                 float* __restrict__ S) {
  const int lane  = threadIdx.x & 31;
  const int gwave = blockIdx.x * 8 + (threadIdx.x >> 5);  // 0 .. 32*128-1
  const int tile  = gwave & 127;          // 16 Mblocks x 8 Nblocks
  const int b     = gwave >> 7;
  const int tm    = (tile >> 3) << 5;     // 32-row block
  const int tn    = (tile & 7) << 6;      // 64-col block

  const float* xb    = xo + (size_t)b * NC * NN;
  const int    col   = lane & 15;
  const int    khalf = (lane >> 4) << 1;  // 0 or 2

  v8f acc[2][4];
#pragma unroll
  for (int i = 0; i < 2; ++i)
#pragma unroll
    for (int j = 0; j < 4; ++j) acc[i][j] = (v8f){};

#pragma unroll 2
  for (int k = 0; k < NC; k += 4) {
    const float* rp0 = xb + (size_t)(k + khalf) * NN + col;
    const float* rp1 = rp0 + NN;
    v2f a[2], bf[4];
#pragma unroll
    for (int i = 0; i < 2; ++i) {
      a[i].x = rp0[tm + i * 16];
      a[i].y = rp1[tm + i * 16];
    }
#pragma unroll
    for (int j = 0; j < 4; ++j) {
      bf[j].x = rp0[tn + j * 16];
      bf[j].y = rp1[tn + j * 16];
    }
#pragma unroll
    for (int i = 0; i < 2; ++i)
#pragma unroll
      for (int j = 0; j < 4; ++j)
        acc[i][j] = __builtin_amdgcn_wmma_f32_16x16x4_f32(
            false, a[i], false, bf[j], (short)0, acc[i][j], false, false);
  }

  float* sb = S + (size_t)b * NN * NN;
  const int rowhi = (lane >> 4) << 3;
#pragma unroll
  for (int i = 0; i < 2; ++i)
#pragma unroll
    for (int r = 0; r < 8; ++r) {
      float* rowp = sb + (size_t)(tm + i * 16 + r + rowhi) * NN + col;
#pragma unroll
      for (int j = 0; j < 4; ++j)
        rowp[tn + j * 16] = acc[i][j][r];
    }
}

// ---------------------------------------------------------------------------
// Kernel 2: sq[b,n] = sum_m S[b,n,m]^2  (one block per row)
// ---------------------------------------------------------------------------
__global__ __launch_bounds__(256) void soga_rowsq(const float* __restrict__ S,
                                                  float* __restrict__ sq) {
  const int row = blockIdx.x;                 // 0 .. B*N-1
  const float* sr = S + (size_t)row * NN;
  float acc = 0.f;
  for (int i = threadIdx.x; i < NN; i += 256) {
    float v = sr[i];
    acc = fmaf(v, v, acc);
  }
  __shared__ float red[256];
  red[threadIdx.x] = acc;
  __syncthreads();
  for (int s = 128; s > 0; s >>= 1) {
    if (threadIdx.x < s) red[threadIdx.x] += red[threadIdx.x + s];
    __syncthreads();
  }
  if (threadIdx.x == 0) sq[row] = red[0];
}

// ---------------------------------------------------------------------------
// Kernel 3: G = S*S (S symmetric), fused epilogue:
//   D[n,m] = sqrt(max(sq[n]+sq[m]-2*G[n,m], eps))
// Symmetry: G[n,m] = sum_k S[k,n]*S[k,m] -> coalesced row-reads of S.
// Same 2x4 register blocking as the Gram kernel.
// ---------------------------------------------------------------------------
__global__ __launch_bounds__(256) void soga_dist(const float* __restrict__ S,
                                                 const float* __restrict__ sq,
                                                 float* __restrict__ D) {
  const int lane  = threadIdx.x & 31;
  const int gwave = blockIdx.x * 8 + (threadIdx.x >> 5);
  const int tile  = gwave & 127;
  const int b     = gwave >> 7;
  const int tm    = (tile >> 3) << 5;
  const int tn    = (tile & 7) << 6;

  const float* sb    = S + (size_t)b * NN * NN;
  const int    col   = lane & 15;
  const int    khalf = (lane >> 4) << 1;

  v8f acc[2][4];
#pragma unroll
  for (int i = 0; i < 2; ++i)
#pragma unroll
    for (int j = 0; j < 4; ++j) acc[i][j] = (v8f){};

#pragma unroll 2
  for (int k = 0; k < NN; k += 4) {
    const float* rp0 = sb + (size_t)(k + khalf) * NN + col;
    const float* rp1 = rp0 + NN;
    v2f a[2], bf[4];
#pragma unroll
    for (int i = 0; i < 2; ++i) {
      a[i].x = rp0[tm + i * 16];
      a[i].y = rp1[tm + i * 16];
    }
#pragma unroll
    for (int j = 0; j < 4; ++j) {
      bf[j].x = rp0[tn + j * 16];
      bf[j].y = rp1[tn + j * 16];
    }
#pragma unroll
    for (int i = 0; i < 2; ++i)
#pragma unroll
      for (int j = 0; j < 4; ++j)
        acc[i][j] = __builtin_amdgcn_wmma_f32_16x16x4_f32(
            false, a[i], false, bf[j], (short)0, acc[i][j], false, false);
  }

  const float* sqb = sq + b * NN;
  float sqc[4];
#pragma unroll
  for (int j = 0; j < 4; ++j) sqc[j] = sqb[tn + j * 16 + col];

  float* db = D + (size_t)b * NN * NN;
  const int rowhi = (lane >> 4) << 3;
#pragma unroll
  for (int i = 0; i < 2; ++i)
#pragma unroll
    for (int r = 0; r < 8; ++r) {
      const int row = tm + i * 16 + r + rowhi;
      const float sqr = sqb[row];
      float* rowp = db + (size_t)row * NN + col;
#pragma unroll
      for (int j = 0; j < 4; ++j) {
        float d2 = sqr + sqc[j] - 2.0f * acc[i][j][r];
        d2 = fmaxf(d2, EPS_SQRT);
        rowp[tn + j * 16] = sqrtf(d2);
      }
    }
}

// ---------------------------------------------------------------------------
// Kernel 4: split-K partial GEMM  part[s,b,o] = sum_{k in split s} D[b,k]*W[k,o]
// 256 threads = 256 output cols; 32 fp32 accumulators (one per batch) per
// thread; D chunk staged in LDS (broadcast reads, conflict-free); W rows read
// coalesced and exactly once across the whole grid (HBM-optimal: 256 MB).
// Next W chunk is prefetched (global_prefetch) while the current one MACs.
// ---------------------------------------------------------------------------
__global__ __launch_bounds__(256) void soga_outpart(const float* __restrict__ D,
                                                    const float* __restrict__ W,
                                                    float* __restrict__ part) {
  const int s  = blockIdx.x;                 // split id
  const int o  = threadIdx.x;                // output column
  const int KC = (NN * NN) / NSPLIT;         // 1024
  const int k0 = s * KC;

  __shared__ float dtile[NB * 32];           // 32 batches x 32 k
  float acc[NB];
#pragma unroll
  for (int b = 0; b < NB; ++b) acc[b] = 0.f;

  for (int kc = 0; kc < KC; kc += 32) {
    __syncthreads();
#pragma unroll
    for (int i = 0; i < 4; ++i) {
      const int flat = threadIdx.x + i * 256;
      const int bb = flat >> 5, kk = flat & 31;
      dtile[flat] = D[(size_t)bb * (NN * NN) + k0 + kc + kk];
    }
    __syncthreads();
    if (kc + 32 < KC)  // prefetch next W chunk while this one computes
      __builtin_prefetch(&W[(size_t)(k0 + kc + 32) * NOUT + o], 0, 1);
    for (int kk = 0; kk < 32; ++kk) {
      const float w = W[(size_t)(k0 + kc + kk) * NOUT + o];
#pragma unroll
      for (int b = 0; b < NB; ++b)
        acc[b] = fmaf(dtile[b * 32 + kk], w, acc[b]);
    }
  }

  float* p = part + (size_t)s * NB * NOUT;
#pragma unroll
  for (int b = 0; b < NB; ++b) p[b * NOUT + o] = acc[b];
}

// ---------------------------------------------------------------------------
// Kernel 5: reduce splits + bias, then per-batch L2 normalize.
// ---------------------------------------------------------------------------
__global__ __launch_bounds__(256) void soga_final(const float* __restrict__ part,
                                                  const float* __restrict__ bias,
                                                  float* __restrict__ out) {
  const int b = blockIdx.x;
  const int o = threadIdx.x;
  float acc = bias[o];
  for (int s = 0; s < NSPLIT; ++s)
    acc += part[(size_t)s * NB * NOUT + b * NOUT + o];

  __shared__ float red[256];
  red[o] = acc * acc;
  __syncthreads();
  for (int st = 128; st > 0; st >>= 1) {
    if (o < st) red[o] += red[o + st];
    __syncthreads();
  }
  const float norm = fmaxf(sqrtf(red[0]), EPS_NORM);
  out[b * NOUT + o] = acc / norm;
}

// ---------------------------------------------------------------------------
// Workspace layout (bytes):
//   S:    [0,            33554432)    32 MB
//   D:    [33554432,     67108864)    32 MB
//   sq:   [67108864,     67174400)    64 KB
//   part: [67174400,     75563008)    8 MB   (NSPLIT*NB*NOUT*4)
// ---------------------------------------------------------------------------
extern "C" void kernel_launch(void* const* d_in, const int* in_sizes, int n_in,
                              void* d_out, int out_size, void* d_ws, size_t ws_size,
                              hipStream_t stream) {
  (void)in_sizes; (void)n_in; (void)out_size; (void)ws_size;
  const float* xo   = (const float*)d_in[2];   // [B, C, N]
  const float* W    = (const float*)d_in[3];   // [N*N, OUT]
  const float* bias = (const float*)d_in[4];   // [OUT]
  float* out = (float*)d_out;                  // [B, OUT]

  char*  ws   = (char*)d_ws;
  float* S    = (float*)(ws);
  float* D    = (float*)(ws + 33554432);
  float* sq   = (float*)(ws + 67108864);
  float* part = (float*)(ws + 67108864 + 65536);

  soga_gram  <<<512,     256, 0, stream>>>(xo, S);      // 4096 waves, 2x4 tiles each
  soga_rowsq <<<NB * NN, 256, 0, stream>>>(S, sq);
  soga_dist  <<<512,     256, 0, stream>>>(S, sq, D);   // 4096 waves
  soga_outpart<<<NSPLIT, 256, 0, stream>>>(D, W, part);
  soga_final <<<NB,      256, 0, stream>>>(part, bias, out);
}